// YoloLoss_18416819765316
// MI455X (gfx1250) — compile-verified
//
#include <hip/hip_runtime.h>
#include <stdint.h>

// ---------------------------------------------------------------------------
// YOLO loss on MI455X (gfx1250). Bandwidth-bound streaming reduction:
// ~174 MB read per call -> ~7.5us roofline at 23.3 TB/s (working set < 192MB L2).
// CDNA5 paths: global_load_async_to_lds_b128 (ASYNCcnt) stages the 120B-per-cell
// pred records into LDS as aligned 128-bit transfers, overlapped with direct
// coalesced b128 loads of the aligned streams; wave32 shfl_xor butterfly +
// 4 global_atomic_add_f32 per block finish the reduction.
// ---------------------------------------------------------------------------

#define CELLS_PER_BLOCK 256
#define PRED_CH 30
#define CLS_CH 20
#define N_BATCH 4096.0f
#define INV_S (1.0f / 14.0f)

// gfx1250 async global->LDS copy, GVS mode: sgpr64 base + per-lane byte offset.
// VDST = 32-bit LDS byte address, VADDR = 32-bit byte offset, SADDR = base.
__device__ __forceinline__ void async_to_lds_b128(unsigned lds_byte,
                                                  unsigned goff_byte,
                                                  unsigned long long gbase) {
  asm volatile("global_load_async_to_lds_b128 %0, %1, %2"
               :
               : "v"(lds_byte), "v"(goff_byte), "s"(gbase)
               : "memory");
}

__device__ __forceinline__ void wait_async_lds() {
  asm volatile("s_wait_asynccnt 0" ::: "memory");
}

__global__ void __launch_bounds__(CELLS_PER_BLOCK) yolo_init_acc(float* acc) {
  if (threadIdx.x < 4) acc[threadIdx.x] = 0.0f;
}

__global__ void __launch_bounds__(CELLS_PER_BLOCK) yolo_partial(
    const float* __restrict__ pred,        // [n_cells, 30]
    const float* __restrict__ tbox,        // [n_cells, 4]
    const float* __restrict__ tcls,        // [n_cells, 20]
    const unsigned char* __restrict__ mask,// [n_cells] bool
    float* __restrict__ acc,               // [4] un-normalized sums
    int n_cells) {
  __shared__ float sp[CELLS_PER_BLOCK * PRED_CH];  // 30720 B staged pred
  __shared__ float sred[8][4];                     // per-wave partials

  const int tid = threadIdx.x;
  const int c0 = blockIdx.x * CELLS_PER_BLOCK;
  const int c = c0 + tid;
  const bool valid = c < n_cells;

  // ---- 1) Kick off async staging of this block's pred chunk into LDS.
  // Chunk byte offset = c0*120 (multiple of 30720 -> 16B aligned vs base).
  {
    const unsigned long long pbase =
        (unsigned long long)(uintptr_t)(pred) + (unsigned long long)c0 * (PRED_CH * 4ull);
    const unsigned lds0 = (unsigned)(uintptr_t)(&sp[0]);  // low 32 bits = LDS offset
    const int cells_here = min(CELLS_PER_BLOCK, n_cells - c0);
    const int n_xfer = (cells_here * PRED_CH * 4) / 16;   // 1920 for a full block
    for (int i = tid; i < n_xfer; i += CELLS_PER_BLOCK) {
      async_to_lds_b128(lds0 + (unsigned)(i * 16), (unsigned)(i * 16), pbase);
    }
  }

  // ---- 2) While the async engine streams pred, issue the direct coalesced
  //         b128 loads of the aligned per-cell streams into registers.
  float4 tb = make_float4(0.f, 0.f, 0.f, 0.f);
  float4 tc0 = tb, tc1 = tb, tc2 = tb, tc3 = tb, tc4 = tb;
  bool mk = false;
  if (valid) {
    tb = *(const float4*)(tbox + (size_t)c * 4);          // 16B/cell, aligned
    const float4* TC4 = (const float4*)(tcls + (size_t)c * CLS_CH);  // 80B/cell
    tc0 = TC4[0]; tc1 = TC4[1]; tc2 = TC4[2]; tc3 = TC4[3]; tc4 = TC4[4];
    mk = mask[c] != 0;
  }

  // ---- 3) Drain this wave's ASYNCcnt, then make all LDS writes visible.
  wait_async_lds();
  __syncthreads();

  // ---- 4) Per-cell loss terms (exactly mirrors the reference math).
  float preg = 0.0f, pconf = 0.0f, pnoobj = 0.0f, pcls = 0.0f;
  if (valid) {
    const float* P = &sp[tid * PRED_CH];
    const float mf = mk ? 1.0f : 0.0f;
    const float nf = 1.0f - mf;

    const float p1x = P[0], p1y = P[1], p1w = P[2], p1h = P[3], p1c = P[4];
    const float p2x = P[5], p2y = P[6], p2w = P[7], p2h = P[8], p2c = P[9];

    // class loss: sum_j (pred_cls[j] - target_cls[j])^2
    float tcv[CLS_CH];
    *(float4*)&tcv[0]  = tc0; *(float4*)&tcv[4]  = tc1; *(float4*)&tcv[8]  = tc2;
    *(float4*)&tcv[12] = tc3; *(float4*)&tcv[16] = tc4;
    float cs = 0.0f;
#pragma unroll
    for (int j = 0; j < CLS_CH; ++j) {
      const float d = P[10 + j] - tcv[j];
      cs = fmaf(d, d, cs);
    }
    pcls = mf * cs;

    // no-object confidence loss
    pnoobj = nf * (p1c * p1c + p2c * p2c);

    // target box -> xyxy
    const float t_x1 = tb.x * INV_S - 0.5f * tb.z;
    const float t_y1 = tb.y * INV_S - 0.5f * tb.w;
    const float t_x2 = tb.x * INV_S + 0.5f * tb.z;
    const float t_y2 = tb.y * INV_S + 0.5f * tb.w;
    const float ta = (t_x2 - t_x1) * (t_y2 - t_y1);

    auto iou_vs_target = [&](float x, float y, float w, float h) -> float {
      const float x1 = x * INV_S - 0.5f * w, y1 = y * INV_S - 0.5f * h;
      const float x2 = x * INV_S + 0.5f * w, y2 = y * INV_S + 0.5f * h;
      const float ltx = fmaxf(x1, t_x1), lty = fmaxf(y1, t_y1);
      const float rbx = fminf(x2, t_x2), rby = fminf(y2, t_y2);
      const float iw = fmaxf(rbx - ltx, 0.0f), ih = fmaxf(rby - lty, 0.0f);
      const float inter = iw * ih;
      const float a = (x2 - x1) * (y2 - y1);
      return inter / (a + ta - inter);
    };

    const float iou1 = iou_vs_target(p1x, p1y, p1w, p1h);
    const float iou2 = iou_vs_target(p2x, p2y, p2w, p2h);
    const bool take1 = iou1 > iou2;
    const float biou = take1 ? iou1 : iou2;
    const float bx = take1 ? p1x : p2x;
    const float by = take1 ? p1y : p2y;
    const float bw = take1 ? p1w : p2w;
    const float bh = take1 ? p1h : p2h;
    const float bc = take1 ? p1c : p2c;

    const float dx = bx - tb.x, dy = by - tb.y;
    const float pwq = sqrtf(mk ? bw : 1.0f);
    const float phq = sqrtf(mk ? bh : 1.0f);
    const float twq = sqrtf(mk ? tb.z : 1.0f);
    const float thq = sqrtf(mk ? tb.w : 1.0f);
    const float dw = pwq - twq, dh = phq - thq;

    preg = mf * (dx * dx + dy * dy + dw * dw + dh * dh);
    const float dc = bc - biou;
    pconf = mf * dc * dc;
  }

  // ---- 5) wave32 butterfly reduction (5 steps), then 8-wave LDS combine.
#pragma unroll
  for (int off = 16; off > 0; off >>= 1) {
    preg   += __shfl_xor(preg, off, 32);
    pconf  += __shfl_xor(pconf, off, 32);
    pnoobj += __shfl_xor(pnoobj, off, 32);
    pcls   += __shfl_xor(pcls, off, 32);
  }
  const int wave = tid >> 5;
  if ((tid & 31) == 0) {
    sred[wave][0] = preg;
    sred[wave][1] = pconf;
    sred[wave][2] = pnoobj;
    sred[wave][3] = pcls;
  }
  __syncthreads();
  if (tid < 4) {
    float s = 0.0f;
#pragma unroll
    for (int w = 0; w < 8; ++w) s += sred[w][tid];
    atomicAdd(&acc[tid], s);  // global_atomic_add_f32, 4 per block
  }
}

__global__ void __launch_bounds__(32) yolo_finalize(const float* __restrict__ acc,
                                                    float* __restrict__ out) {
  if (threadIdx.x == 0) {
    const float invN = 1.0f / N_BATCH;
    const float reg   = acc[0] * invN;
    const float conf  = acc[1] * invN;
    const float noobj = acc[2] * invN;
    const float cls   = acc[3] * invN;
    out[0] = 5.0f * reg + conf + 0.5f * noobj + cls;  // total
    out[1] = reg;
    out[2] = conf;
    out[3] = noobj;
    out[4] = cls;
  }
}

extern "C" void kernel_launch(void* const* d_in, const int* in_sizes, int n_in,
                              void* d_out, int out_size, void* d_ws, size_t ws_size,
                              hipStream_t stream) {
  const float* pred         = (const float*)d_in[0];
  const float* tbox         = (const float*)d_in[1];
  const float* tcls         = (const float*)d_in[2];
  const unsigned char* mask = (const unsigned char*)d_in[3];
  float* out = (float*)d_out;
  float* acc = (float*)d_ws;  // 4 floats of scratch

  const int n_cells = in_sizes[3];                       // 802816 = 3136*256
  const int blocks = (n_cells + CELLS_PER_BLOCK - 1) / CELLS_PER_BLOCK;

  yolo_init_acc<<<1, 32, 0, stream>>>(acc);
  yolo_partial<<<blocks, CELLS_PER_BLOCK, 0, stream>>>(pred, tbox, tcls, mask,
                                                       acc, n_cells);
  yolo_finalize<<<1, 32, 0, stream>>>(acc, out);
}